// ImproveFFBlock3_69715909149113
// MI455X (gfx1250) — compile-verified
//
#include <hip/hip_runtime.h>

// ---------------------------------------------------------------------------
// Shapes (from the reference):
//   x1 : (8,128,128,128) f32   -> 2x2 pool -> tile x2   over channels
//   x2 : (8, 32,256,256) f32   -> 4x4 pool -> tile x8
//   x3 : (8,  2,512,512) f32   -> 8x8 pool -> tile x128
//   ff : (8,256, 64, 64) f32
//   out = relu(y1+y2+y3+ff) : (8,256,64,64) f32
//
// Bandwidth-bound: ~208 MiB minimum HBM traffic -> ~9.4us @ 23.3 TB/s.
// Strategy: pre-pool x3 (16 MiB -> 256 KiB, L2-resident) using the CDNA5
// async LDS data mover; then one fused kernel that reads x1/x2/ff exactly
// once with NT-hinted wide loads and writes out with NT stores.
// ---------------------------------------------------------------------------

typedef float v2f __attribute__((ext_vector_type(2)));
typedef float v4f __attribute__((ext_vector_type(4)));
typedef int   v4i __attribute__((ext_vector_type(4)));

#if defined(__gfx1250__) &&                                                   \
    __has_builtin(__builtin_amdgcn_global_load_async_to_lds_b128) &&          \
    __has_builtin(__builtin_amdgcn_s_wait_asynccnt)
#define USE_ASYNC_LDS 1
#else
#define USE_ASYNC_LDS 0
#endif

__device__ __forceinline__ v4f nt4(const float* p) {
    return __builtin_nontemporal_load((const v4f*)p);
}
__device__ __forceinline__ v2f nt2(const float* p) {
    return __builtin_nontemporal_load((const v2f*)p);
}
__device__ __forceinline__ v4f vmax4(v4f a, v4f b) {
    v4f r;
    r.x = fmaxf(a.x, b.x); r.y = fmaxf(a.y, b.y);
    r.z = fmaxf(a.z, b.z); r.w = fmaxf(a.w, b.w);
    return r;
}
__device__ __forceinline__ float hmax4(v4f v) {
    return fmaxf(fmaxf(v.x, v.y), fmaxf(v.z, v.w));
}

// ---------------------------------------------------------------------------
// Kernel P: 8x8 max-pool of x3 (8,2,512,512) -> pool3 (8,2,64,64) in d_ws.
// Block = 256 threads = 4 pooled rows x 64 pooled cols for one (b,c3,hblk).
// Input slab per block: 32 rows x 512 cols = 64 KiB, CONTIGUOUS in memory ->
// staged into LDS with the async data mover, then reduced via ds_load_b128.
// Grid: 8*2*16 = 256 blocks.
// ---------------------------------------------------------------------------
__global__ __launch_bounds__(256) void pool_x3_kernel(
    const float* __restrict__ x3, float* __restrict__ pool3) {
    const int tid  = threadIdx.x;
    const int hl   = tid >> 6;        // 0..3  pooled row within block
    const int w    = tid & 63;        // 0..63 pooled col
    const int hblk = blockIdx.x & 15; // 16 blocks of 4 pooled rows
    const int bc   = blockIdx.x >> 4; // b*2 + c3
    const int h    = hblk * 4 + hl;   // pooled row 0..63

    float m;
#if USE_ASYNC_LDS
    __shared__ float lds[32 * 512];   // 64 KiB slab
    const float* gbase = x3 + ((size_t)bc * 512 + (size_t)hblk * 32) * 512;
    // 64 KiB = 16 iterations x 256 lanes x 16B, perfectly coalesced.
#pragma unroll
    for (int it = 0; it < 16; ++it) {
        const int fo = it * 1024 + tid * 4; // float offset
        __builtin_amdgcn_global_load_async_to_lds_b128(
            (__attribute__((address_space(1))) v4i*)(gbase + fo),
            (__attribute__((address_space(3))) v4i*)(lds + fo),
            /*offset=*/0, /*cpol=*/0);
    }
    __builtin_amdgcn_s_wait_asynccnt(0);
    __syncthreads();

    const float* lrow = lds + (size_t)(hl * 8) * 512 + (size_t)w * 8;
    v4f acc = vmax4(*(const v4f*)(lrow), *(const v4f*)(lrow + 4));
#pragma unroll
    for (int r = 1; r < 8; ++r) {
        const float* p = lrow + (size_t)r * 512;
        acc = vmax4(acc, vmax4(*(const v4f*)(p), *(const v4f*)(p + 4)));
    }
    m = hmax4(acc);
#else
    const float* gb = x3 + ((size_t)bc * 512 + (size_t)h * 8) * 512 + (size_t)w * 8;
    v4f acc = vmax4(nt4(gb), nt4(gb + 4));
#pragma unroll
    for (int r = 1; r < 8; ++r) {
        const float* p = gb + (size_t)r * 512;
        acc = vmax4(acc, vmax4(nt4(p), nt4(p + 4)));
    }
    m = hmax4(acc);
#endif
    pool3[((size_t)bc * 64 + h) * 64 + w] = m;
}

// ---------------------------------------------------------------------------
// Kernel F: fused pools + tiled add + relu.
// Thread = (b, c0 in [0,32), h, w); produces 8 output channels c0+32k.
//  - x1 pooled (2x2) for c1 = c0+32j, j=0..3  (each x1 element read once)
//  - x2 pooled (4x4) for c2 = c0              (each x2 element read once)
//  - x3 pooled value from d_ws (L2-resident 256 KiB), or direct if !USE_WS
// Grid: 8*32*16 = 4096 blocks x 256 threads.
// ---------------------------------------------------------------------------
template <bool USE_WS>
__global__ __launch_bounds__(256) void fused_kernel(
    const float* __restrict__ x1, const float* __restrict__ x2,
    const float* __restrict__ x3, const float* __restrict__ ff,
    const float* __restrict__ pool3, float* __restrict__ out) {
    const int tid  = threadIdx.x;
    const int hl   = tid >> 6;
    const int w    = tid & 63;
    const int hblk = blockIdx.x & 15;
    const int t    = blockIdx.x >> 4;
    const int c0   = t & 31;
    const int b    = t >> 5;
    const int h    = hblk * 4 + hl;

    // ---- pooled x3 contribution (channel c0 & 1) ----
    float p3;
    if (USE_WS) {
        p3 = pool3[(((size_t)b * 2 + (c0 & 1)) * 64 + h) * 64 + w]; // RT: stays in L2
    } else {
        const float* gb =
            x3 + (((size_t)b * 2 + (c0 & 1)) * 512 + (size_t)h * 8) * 512 + (size_t)w * 8;
        v4f acc = vmax4(nt4(gb), nt4(gb + 4));
#pragma unroll
        for (int r = 1; r < 8; ++r) {
            const float* p = gb + (size_t)r * 512;
            acc = vmax4(acc, vmax4(nt4(p), nt4(p + 4)));
        }
        p3 = hmax4(acc);
    }

    // ---- pooled x2 (4x4), channel c0 ----
    const float* g2 =
        x2 + (((size_t)b * 32 + c0) * 256 + (size_t)h * 4) * 256 + (size_t)w * 4;
    v4f a2 = nt4(g2);
    a2 = vmax4(a2, nt4(g2 + 256));
    a2 = vmax4(a2, nt4(g2 + 512));
    a2 = vmax4(a2, nt4(g2 + 768));
    const float p2 = hmax4(a2);

    // ---- pooled x1 (2x2), channels c0 + 32j ----
    float p1[4];
#pragma unroll
    for (int j = 0; j < 4; ++j) {
        const int c1 = c0 + 32 * j;
        const float* g1 =
            x1 + (((size_t)b * 128 + c1) * 128 + (size_t)h * 2) * 128 + (size_t)w * 2;
        const v2f r0 = nt2(g1);
        const v2f r1 = nt2(g1 + 128);
        p1[j] = fmaxf(fmaxf(r0.x, r0.y), fmaxf(r1.x, r1.y));
    }

    // ---- combine: out[c0+32k] = relu(p1[k%4] + p2 + p3 + ff) ----
    const float base = p2 + p3;
#pragma unroll
    for (int k = 0; k < 8; ++k) {
        const int c = c0 + 32 * k;
        const size_t idx = (((size_t)b * 256 + c) * 64 + h) * 64 + w;
        const float v = p1[k & 3] + base + __builtin_nontemporal_load(ff + idx);
        __builtin_nontemporal_store(fmaxf(v, 0.0f), out + idx);
    }
}

// ---------------------------------------------------------------------------
extern "C" void kernel_launch(void* const* d_in, const int* in_sizes, int n_in,
                              void* d_out, int out_size, void* d_ws, size_t ws_size,
                              hipStream_t stream) {
    const float* x1 = (const float*)d_in[0]; // (8,128,128,128)
    const float* x2 = (const float*)d_in[1]; // (8, 32,256,256)
    const float* x3 = (const float*)d_in[2]; // (8,  2,512,512)
    const float* ff = (const float*)d_in[3]; // (8,256, 64, 64)
    float* out      = (float*)d_out;         // (8,256, 64, 64)
    float* pool3    = (float*)d_ws;          // 8*2*64*64 floats = 256 KiB

    const size_t ws_needed = (size_t)8 * 2 * 64 * 64 * sizeof(float);
    const bool use_ws = (ws_size >= ws_needed);

    if (use_ws) {
        pool_x3_kernel<<<256, 256, 0, stream>>>(x3, pool3);
        fused_kernel<true><<<8 * 32 * 16, 256, 0, stream>>>(x1, x2, x3, ff, pool3, out);
    } else {
        // Deterministic fallback (x3 pooled redundantly in-kernel).
        fused_kernel<false><<<8 * 32 * 16, 256, 0, stream>>>(x1, x2, x3, ff, pool3, out);
    }
}